// HyperbolicKuramoto_7352984010850
// MI455X (gfx1250) — compile-verified
//
#include <hip/hip_runtime.h>
#include <math.h>

// ---------------------------------------------------------------------------
// Hyperbolic Kuramoto, N=8192, 50 steps.
// Since K and omega are real:  S = K@conj(z) = u - i v,  T = K@z = u + i v
// with u = K@z_re, v = K@z_im  (two real fp32 matvecs per step).
// Matvecs done with V_WMMA_F32_16X16X4_F32 (exact fp32 FMA path).
// ---------------------------------------------------------------------------

typedef float v2f __attribute__((ext_vector_type(2)));
typedef float v4f __attribute__((ext_vector_type(4)));
typedef float v8f __attribute__((ext_vector_type(8)));

#define NN 8192
#define KSPLIT 8                       // K-dim split for occupancy + determinism
#define ITERS (NN / 8)                 // 1024 iterations of 8 k-values each
#define ITERS_PER_SEG (ITERS / KSPLIT) // 128
#define PANEL_FLOATS (ITERS * 128)     // per-iter: 32 lanes x 4 slots

// Deinterleave z[N,2] -> z_re[N], z_im[N]
__global__ void hk_init(const float* __restrict__ z,
                        float* __restrict__ zre, float* __restrict__ zim) {
  int m = blockIdx.x * blockDim.x + threadIdx.x;
  if (m < NN) { zre[m] = z[2 * m]; zim[m] = z[2 * m + 1]; }
}

// Build the WMMA B-operand panel for the current z.
// For 8-k iteration i, lane l, slot s (s = the lane's s-th float of a b128):
//   half = l>>4, n = l&15, k = 8*i + 4*half + s
//   value = (n==0) ? z_re[k] : (n==1) ? z_im[k] : 0
// WMMA#1 consumes slots (0,1) as B vgpr0/1, WMMA#2 consumes slots (2,3).
// This matches the 16x16x4 layout: vgpr0 lower lanes = k-slot0, vgpr1 lower =
// slot1, vgpr0 upper = slot2, vgpr1 upper = slot3 (mirrors the A layout).
__global__ void hk_prep(const float* __restrict__ zre,
                        const float* __restrict__ zim,
                        float* __restrict__ panel) {
  int idx = blockIdx.x * blockDim.x + threadIdx.x;  // 0 .. PANEL_FLOATS-1
  int i    = idx >> 7;
  int r    = idx & 127;
  int lane = r >> 2;
  int s    = r & 3;
  int n    = lane & 15;
  int half = lane >> 4;
  int k    = 8 * i + 4 * half + s;
  float val = 0.0f;
  if (n == 0)      val = zre[k];
  else if (n == 1) val = zim[k];
  panel[idx] = val;
}

// One wave = (row tile of 16, K segment of 1024). Accumulates
// C[16,16] += K_tile[16,4] * B[4,16] with B columns 0/1 = z_re/z_im.
__global__ void
__launch_bounds__(256)
hk_matvec(const float* __restrict__ K, const float* __restrict__ panel,
          float* __restrict__ upart, float* __restrict__ vpart) {
  const int lane = threadIdx.x & 31;
  const int seg  = threadIdx.x >> 5;   // 8 waves per block = 8 segments
  const int tile = blockIdx.x;         // 512 row tiles
  const int n    = lane & 15;
  const int half = lane >> 4;
  const int row  = tile * 16 + n;

  // Per lane, iteration i reads K[row][k0 + 4*half .. +3], k0 = seg*1024 + 8*i
  const v4f* __restrict__ kp =
      (const v4f*)(K + (size_t)row * NN + (size_t)seg * (NN / KSPLIT) + 4 * half);
  const v4f* __restrict__ pp =
      (const v4f*)(panel + (size_t)seg * ITERS_PER_SEG * 128 + lane * 4);

  v8f acc = {};
#pragma unroll 4
  for (int i = 0; i < ITERS_PER_SEG; ++i) {
    v4f a = kp[2 * i];    // K[row][k0+4h .. k0+4h+3]
    v4f b = pp[32 * i];   // matching z panel slots
    v2f a0 = {a.x, a.y}, a1 = {a.z, a.w};
    v2f b0 = {b.x, b.y}, b1 = {b.z, b.w};
    // D = A(16x4 f32) * B(4x16 f32) + C, exact fp32
    acc = __builtin_amdgcn_wmma_f32_16x16x4_f32(false, a0, false, b0,
                                                (short)0, acc, false, false);
    acc = __builtin_amdgcn_wmma_f32_16x16x4_f32(false, a1, false, b1,
                                                (short)0, acc, false, false);
  }

  // C layout: vgpr p -> M=p (lanes 0-15) / M=p+8 (lanes 16-31); lane%16 = N.
  // Column 0 = u, column 1 = v.
  if (n < 2) {
    float* dst = (n == 0 ? upart : vpart) + (size_t)seg * NN + tile * 16 + 8 * half;
#pragma unroll
    for (int p = 0; p < 8; ++p) dst[p] = acc[p];
  }
}

// Deterministic segment reduction + complex Euler step + 0.999 clamp.
__global__ void hk_update(const float* __restrict__ upart,
                          const float* __restrict__ vpart,
                          const float* __restrict__ zre_in,
                          const float* __restrict__ zim_in,
                          const float* __restrict__ omega,
                          const float* __restrict__ dtp,
                          float* __restrict__ zre_out,
                          float* __restrict__ zim_out) {
  int m = blockIdx.x * blockDim.x + threadIdx.x;
  if (m >= NN) return;
  float u = 0.0f, v = 0.0f;
#pragma unroll
  for (int s = 0; s < KSPLIT; ++s) {
    u += upart[(size_t)s * NN + m];
    v += vpart[(size_t)s * NN + m];
  }
  const float inv2N = 1.0f / (2.0f * (float)NN);
  float zr = zre_in[m], zi = zim_in[m];
  float z2r = zr * zr - zi * zi;     // z^2
  float z2i = 2.0f * zr * zi;
  float w  = omega[m];
  float dt = *dtp;
  // dz = -inv2N * (u - i v) * z^2 + omega*z + inv2N*(u + i v)
  float dzr = -inv2N * (u * z2r + v * z2i) + w * zr + inv2N * u;
  float dzi = -inv2N * (u * z2i - v * z2r) + w * zi + inv2N * v;
  zr += dt * dzr;
  zi += dt * dzi;
  float a = sqrtf(zr * zr + zi * zi);
  if (a >= 0.999f) {
    float sc = 0.999f / (a + 1e-8f);
    zr *= sc; zi *= sc;
  }
  zre_out[m] = zr;
  zim_out[m] = zi;
}

// Final norm clamp + interleave to [N,2]
__global__ void hk_final(const float* __restrict__ zre,
                         const float* __restrict__ zim,
                         float* __restrict__ out) {
  int m = blockIdx.x * blockDim.x + threadIdx.x;
  if (m >= NN) return;
  float zr = zre[m], zi = zim[m];
  float a = sqrtf(zr * zr + zi * zi);
  if (a >= 0.999f) {
    float sc = 0.999f / (a + 1e-8f);
    zr *= sc; zi *= sc;
  }
  out[2 * m]     = zr;
  out[2 * m + 1] = zi;
}

extern "C" void kernel_launch(void* const* d_in, const int* in_sizes, int n_in,
                              void* d_out, int out_size, void* d_ws, size_t ws_size,
                              hipStream_t stream) {
  (void)in_sizes; (void)n_in; (void)out_size; (void)ws_size;
  const float* z     = (const float*)d_in[0];  // [N,2]
  const float* K     = (const float*)d_in[1];  // [N,N]
  const float* omega = (const float*)d_in[2];  // [N]
  const float* dt    = (const float*)d_in[3];  // scalar
  // d_in[4] = steps (50, fixed by the reference)

  float* ws = (float*)d_ws;
  float* zreA  = ws;                 // N
  float* zimA  = ws + NN;            // N
  float* zreB  = ws + 2 * NN;        // N
  float* zimB  = ws + 3 * NN;        // N
  float* panel = ws + 4 * NN;                  // PANEL_FLOATS (512 KB)
  float* upart = panel + PANEL_FLOATS;         // KSPLIT * N
  float* vpart = upart + (size_t)KSPLIT * NN;  // KSPLIT * N

  float* zre[2] = {zreA, zreB};
  float* zim[2] = {zimA, zimB};

  hk_init<<<NN / 256, 256, 0, stream>>>(z, zreA, zimA);

  int cur = 0;
  for (int step = 0; step < 50; ++step) {
    hk_prep<<<PANEL_FLOATS / 256, 256, 0, stream>>>(zre[cur], zim[cur], panel);
    hk_matvec<<<NN / 16, 256, 0, stream>>>(K, panel, upart, vpart);
    hk_update<<<NN / 256, 256, 0, stream>>>(upart, vpart, zre[cur], zim[cur],
                                            omega, dt, zre[1 - cur], zim[1 - cur]);
    cur ^= 1;
  }

  hk_final<<<NN / 256, 256, 0, stream>>>(zre[cur], zim[cur], (float*)d_out);
}